// SurfaceLoss_48679159333040
// MI455X (gfx1250) — compile-verified
//
#include <hip/hip_runtime.h>
#include <math.h>

// Surface loss: boundary-distance-weighted BCE, reduced to a scalar.
// EDT phase 2 (min-plus with quadratic kernel) mapped onto V_WMMA_F32_16X16X4_F32.
// Row staging into LDS done via the Tensor Data Mover (tensor_load_to_lds).

#define EDT_INF 1000000.0f
#define LAMW    0.5f
#define HDIM    384
#define WDIM    384

typedef __attribute__((ext_vector_type(2))) float v2f;
typedef __attribute__((ext_vector_type(8))) float v8f;
typedef unsigned int u32x4 __attribute__((ext_vector_type(4)));
typedef int          i32x8 __attribute__((ext_vector_type(8)));
typedef int          i32x4 __attribute__((ext_vector_type(4)));

// ---------------------------------------------------------------------------
// Kernel 0: zero the per-image max slots.
// ---------------------------------------------------------------------------
__global__ void k_init_max(unsigned* mb, int nB) {
    int i = threadIdx.x;
    if (i < nB) mb[i] = 0u;
}

// ---------------------------------------------------------------------------
// Kernel 1: exact 1D column EDT via two cummin scans (both polarities).
// Writes g^2 (clamped at EDT_INF before squaring), layout [B,H,W].
// One thread per (b, x) column; consecutive threads -> consecutive x (coalesced).
// ---------------------------------------------------------------------------
__global__ __launch_bounds__(256)
void k_edt_cols(const float* __restrict__ t,
                float* __restrict__ g2a, float* __restrict__ g2b, int nB) {
    int col = blockIdx.x * blockDim.x + threadIdx.x;
    if (col >= nB * WDIM) return;
    int b = col / WDIM, x = col - b * WDIM;
    size_t base = (size_t)b * HDIM * WDIM + x;
    const float* tp = t + base;
    float* pa = g2a + base;
    float* pb = g2b + base;

    // forward cummin of (a - y); store fwd+y
    float ma = 1.0e30f, mb = 1.0e30f;
    for (int y = 0; y < HDIM; ++y) {
        float tv = tp[(size_t)y * WDIM];
        float aa = tv > 0.5f ? EDT_INF : 0.0f;   // edt(t)
        float ab = tv > 0.5f ? 0.0f : EDT_INF;   // edt(1-t)
        float fy = (float)y;
        ma = fminf(ma, aa - fy);
        mb = fminf(mb, ab - fy);
        pa[(size_t)y * WDIM] = ma + fy;
        pb[(size_t)y * WDIM] = mb + fy;
    }
    // backward cummin of (a + y); combine, clamp, square
    ma = 1.0e30f; mb = 1.0e30f;
    for (int y = HDIM - 1; y >= 0; --y) {
        float tv = tp[(size_t)y * WDIM];
        float aa = tv > 0.5f ? EDT_INF : 0.0f;
        float ab = tv > 0.5f ? 0.0f : EDT_INF;
        float fy = (float)y;
        ma = fminf(ma, aa + fy);
        mb = fminf(mb, ab + fy);
        float ga = fminf(fminf(pa[(size_t)y * WDIM], ma - fy), EDT_INF);
        float gb = fminf(fminf(pb[(size_t)y * WDIM], mb - fy), EDT_INF);
        pa[(size_t)y * WDIM] = ga * ga;
        pb[(size_t)y * WDIM] = gb * gb;
    }
}

// ---------------------------------------------------------------------------
// TDM helper: build D# group0 for a 1D row load into LDS.
//   group0: [1:0]=count(1), [63:32]=lds_addr, [120:64]=global_addr, [127:126]=2
// ---------------------------------------------------------------------------
__device__ __forceinline__ u32x4 tdm_group0(unsigned lds_addr,
                                            unsigned long long gaddr) {
    u32x4 g;
    g.x = 1u;                                  // count=1, user descriptor
    g.y = lds_addr;                            // lds_addr [63:32]
    g.z = (unsigned)(gaddr & 0xffffffffull);   // global_addr [95:64]
    g.w = (unsigned)((gaddr >> 32) & 0x01ffffffull) | (2u << 30); // [120:96]|type=2
    return g;
}

// ---------------------------------------------------------------------------
// Kernel 2: row min-plus via WMMA.
//   M[i,j] = (xi - xj)^2 + g2[j] = xi*(-2xj) + 1*(xj^2 + g2[j]) + xi^2
//   A(16x4) = [xi, 1, 0, 0]   B(4x16) = [-2xj ; xj^2+g2 ; 0 ; 0]   C = xi^2
// One block per (b,y) row. Row staged into LDS by the Tensor Data Mover.
// dist overwrites g2a in place; per-image max via integer atomicMax
// (dist >= 0 so the float bit pattern is order-preserving).
// ---------------------------------------------------------------------------
__global__ __launch_bounds__(256)
void k_edt_rows_wmma(float* __restrict__ g2a, const float* __restrict__ g2b,
                     unsigned* __restrict__ maxbits) {
    __shared__ float s_a[WDIM];
    __shared__ float s_b[WDIM];
    __shared__ float s_wmax[8];

    int row  = blockIdx.x;           // b*H + y
    int bimg = row / HDIM;
    size_t base = (size_t)row * WDIM;
    int tid = threadIdx.x;

#if __has_builtin(__builtin_amdgcn_tensor_load_to_lds) && \
    __has_builtin(__builtin_amdgcn_s_wait_tensorcnt)
    // --- TDM staging: two 384x1 f32 tiles, one TDM op each (wave 0 issues) ---
    if (tid < 32) {
        // AS(3)->generic cast puts the LDS byte offset in the low 32 bits.
        unsigned lds_a = (unsigned)(size_t)&s_a[0];
        unsigned lds_b = (unsigned)(size_t)&s_b[0];
        unsigned long long ga = (unsigned long long)(size_t)(g2a + base);
        unsigned long long gb = (unsigned long long)(size_t)(g2b + base);
        // group1: wg_mask=0, data_size=2 (4B), no barrier/iterate/pad,
        // tensor_dim0=384, tensor_dim1=1, tile_dim0=384, tile_dim1=1,
        // tile_dim2=0, tensor_dim0_stride=384, tensor_dim1_stride=0.
        i32x8 g1;
        g1[0] = (int)(2u << 16);           // data_size = 4 bytes
        g1[1] = (int)(384u << 16);         // tensor_dim0[15:0] in bits[63:48]
        g1[2] = (int)(1u << 16);           // tensor_dim0[31:16]=0 | tensor_dim1[15:0]
        g1[3] = (int)(384u << 16);         // tensor_dim1[31:16]=0 | tile_dim0
        g1[4] = (int)1;                    // tile_dim1=1 | tile_dim2=0
        g1[5] = (int)384;                  // tensor_dim0_stride[31:0]
        g1[6] = 0;                         // stride0[47:32]=0 | stride1[15:0]=0
        g1[7] = 0;                         // tensor_dim1_stride[47:16]=0
        i32x4 z4; z4[0] = 0; z4[1] = 0; z4[2] = 0; z4[3] = 0;
        i32x8 z8; z8[0] = 0; z8[1] = 0; z8[2] = 0; z8[3] = 0;
        z8[4] = 0; z8[5] = 0; z8[6] = 0; z8[7] = 0;
        // clang-23 / therock-10.0 form: 6 args (extra i32x8 before cpol)
        __builtin_amdgcn_tensor_load_to_lds(tdm_group0(lds_a, ga), g1,
                                            z4, z4, z8, 0);
        __builtin_amdgcn_tensor_load_to_lds(tdm_group0(lds_b, gb), g1,
                                            z4, z4, z8, 0);
        __builtin_amdgcn_s_wait_tensorcnt(0);   // data in LDS before barrier
    }
    __syncthreads();
#else
    for (int i = tid; i < WDIM; i += 256) {
        s_a[i] = g2a[base + i];
        s_b[i] = g2b[base + i];
    }
    __syncthreads();
#endif

    int wave = tid >> 5;
    int lane = tid & 31;
    int hh   = lane >> 4;   // 16-lane half (K=2,3 slots of A/B are zero)
    int l16  = lane & 15;

    float lmax = 0.0f;

    for (int k = 0; k < 3; ++k) {                 // 8 waves x 3 = 24 tile-rows
        int x0 = (wave * 3 + k) * 16;
        float xi = (float)(x0 + l16);

        v2f A;
        A.x = hh ? 0.0f : xi;
        A.y = hh ? 0.0f : 1.0f;

        v8f C;
#pragma unroll
        for (int v = 0; v < 8; ++v) {
            float xr = (float)(x0 + v + 8 * hh);  // D row M = v + 8*half
            C[v] = xr * xr;
        }

        v8f dacc[2];
#pragma unroll
        for (int p = 0; p < 2; ++p) {
            const float* srow = p ? s_b : s_a;
            v8f dmin;
#pragma unroll
            for (int v = 0; v < 8; ++v) dmin[v] = 3.0e38f;

#pragma unroll 4
            for (int J = 0; J < 24; ++J) {
                int   xj   = J * 16 + l16;
                float xjf  = (float)xj;
                float g2v  = srow[xj];
                v2f Bv;
                Bv.x = hh ? 0.0f : (-2.0f * xjf);
                Bv.y = hh ? 0.0f : (xjf * xjf + g2v);
                v8f D = __builtin_amdgcn_wmma_f32_16x16x4_f32(
                    false, A, false, Bv, (short)0, C, false, false);
#pragma unroll
                for (int v = 0; v < 8; ++v) dmin[v] = fminf(dmin[v], D[v]);
            }
            // min across the 16 lanes of each half (N dimension)
#pragma unroll
            for (int off = 8; off >= 1; off >>= 1) {
#pragma unroll
                for (int v = 0; v < 8; ++v)
                    dmin[v] = fminf(dmin[v], __shfl_xor(dmin[v], off, 32));
            }
            dacc[p] = dmin;
        }

        float dist[8];
#pragma unroll
        for (int v = 0; v < 8; ++v) {
            dist[v] = sqrtf(dacc[0][v]) + sqrtf(dacc[1][v]);
            lmax = fmaxf(lmax, dist[v]);
        }
        if (l16 == 0) {   // lane 0 holds rows x0..x0+7, lane 16 rows x0+8..x0+15
#pragma unroll
            for (int v = 0; v < 8; ++v)
                g2a[base + x0 + 8 * hh + v] = dist[v];
        }
    }

    // block max -> one atomicMax per block (deterministic: integer max)
#pragma unroll
    for (int off = 16; off >= 1; off >>= 1)
        lmax = fmaxf(lmax, __shfl_xor(lmax, off, 32));
    if (lane == 0) s_wmax[wave] = lmax;
    __syncthreads();
    if (tid == 0) {
        float m = s_wmax[0];
        for (int w = 1; w < 8; ++w) m = fmaxf(m, s_wmax[w]);
        atomicMax(maxbits + bimg, __float_as_uint(m));
    }
}

// ---------------------------------------------------------------------------
// Kernel 3: weighted BCE partial sums (fixed-order tree reduce, deterministic).
// ---------------------------------------------------------------------------
__global__ __launch_bounds__(256)
void k_loss_partial(const float* __restrict__ logits,
                    const float* __restrict__ targets,
                    const float* __restrict__ dist,
                    const unsigned* __restrict__ maxbits,
                    float* __restrict__ partials, int n) {
    __shared__ float s_red[8];
    const int HW = HDIM * WDIM;
    float acc = 0.0f;
    for (int i = blockIdx.x * blockDim.x + threadIdx.x; i < n;
         i += gridDim.x * blockDim.x) {
        float z  = logits[i];
        float y  = targets[i];
        float dv = dist[i];
        float mx = __uint_as_float(maxbits[i / HW]);
        float w  = 1.0f + LAMW * dv / (mx + 1e-8f);
        float bce = fmaxf(z, 0.0f) - z * y + log1pf(expf(-fabsf(z)));
        acc += w * bce;
    }
#pragma unroll
    for (int off = 16; off >= 1; off >>= 1) acc += __shfl_xor(acc, off, 32);
    int wave = threadIdx.x >> 5, lane = threadIdx.x & 31;
    if (lane == 0) s_red[wave] = acc;
    __syncthreads();
    if (threadIdx.x == 0) {
        float t = 0.0f;
        for (int w = 0; w < 8; ++w) t += s_red[w];
        partials[blockIdx.x] = t;
    }
}

__global__ __launch_bounds__(256)
void k_finalize(const float* __restrict__ partials, int nparts,
                float* __restrict__ out, float invN) {
    __shared__ float s_red[8];
    float acc = 0.0f;
    for (int i = threadIdx.x; i < nparts; i += 256) acc += partials[i];
#pragma unroll
    for (int off = 16; off >= 1; off >>= 1) acc += __shfl_xor(acc, off, 32);
    int wave = threadIdx.x >> 5, lane = threadIdx.x & 31;
    if (lane == 0) s_red[wave] = acc;
    __syncthreads();
    if (threadIdx.x == 0) {
        float t = 0.0f;
        for (int w = 0; w < 8; ++w) t += s_red[w];
        out[0] = t * invN;
    }
}

// ---------------------------------------------------------------------------
extern "C" void kernel_launch(void* const* d_in, const int* in_sizes, int n_in,
                              void* d_out, int out_size, void* d_ws, size_t ws_size,
                              hipStream_t stream) {
    const float* logits  = (const float*)d_in[0];
    const float* targets = (const float*)d_in[1];
    int n  = in_sizes[0];                 // B*1*H*W
    int nB = n / (HDIM * WDIM);

    // workspace layout: [g2a|dist: n] [g2b: n] [maxbits: 64] [partials]
    float*    g2a      = (float*)d_ws;
    float*    g2b      = g2a + n;
    unsigned* maxbits  = (unsigned*)(g2b + n);
    float*    partials = (float*)(maxbits + 64);

    k_init_max<<<1, 64, 0, stream>>>(maxbits, nB);

    int cols = nB * WDIM;
    k_edt_cols<<<(cols + 255) / 256, 256, 0, stream>>>(targets, g2a, g2b, nB);

    k_edt_rows_wmma<<<nB * HDIM, 256, 0, stream>>>(g2a, g2b, maxbits);

    int nblocks = (n + 1023) / 1024;
    if (nblocks > 4608) nblocks = 4608;
    k_loss_partial<<<nblocks, 256, 0, stream>>>(logits, targets, g2a, maxbits,
                                                partials, n);
    k_finalize<<<1, 256, 0, stream>>>(partials, nblocks, (float*)d_out,
                                      1.0f / (float)n);
}